// DMRU2INetworkLayer_26508538150959
// MI455X (gfx1250) — compile-verified
//
#include <hip/hip_runtime.h>
#include <math.h>

typedef float v2f __attribute__((ext_vector_type(2)));
typedef float v8f __attribute__((ext_vector_type(8)));

#define S_LEN 200
#define S_PAD 208
#define DDIM  128
#define HDIM  128
#define NNEG  64
#define BATCH 1024
#define EPSN  1e-12f
#define MTILES 13

// ---- DPP16 xor-butterfly reduction within each 16-lane group ----
// masks {1,2,7,15} generate the full 16-element xor group.
__device__ __forceinline__ float dpp_add16(float v) {
    int t;
    t = __builtin_amdgcn_update_dpp(0, __float_as_int(v), 0xB1, 0xf, 0xf, true);  // quad_perm xor1
    v += __int_as_float(t);
    t = __builtin_amdgcn_update_dpp(0, __float_as_int(v), 0x4E, 0xf, 0xf, true);  // quad_perm xor2
    v += __int_as_float(t);
    t = __builtin_amdgcn_update_dpp(0, __float_as_int(v), 0x141, 0xf, 0xf, true); // row_half_mirror xor7
    v += __int_as_float(t);
    t = __builtin_amdgcn_update_dpp(0, __float_as_int(v), 0x140, 0xf, 0xf, true); // row_mirror xor15
    v += __int_as_float(t);
    return v;
}
__device__ __forceinline__ float dpp_max16(float v) {
    int t;
    t = __builtin_amdgcn_update_dpp(0, __float_as_int(v), 0xB1, 0xf, 0xf, true);
    v = fmaxf(v, __int_as_float(t));
    t = __builtin_amdgcn_update_dpp(0, __float_as_int(v), 0x4E, 0xf, 0xf, true);
    v = fmaxf(v, __int_as_float(t));
    t = __builtin_amdgcn_update_dpp(0, __float_as_int(v), 0x141, 0xf, 0xf, true);
    v = fmaxf(v, __int_as_float(t));
    t = __builtin_amdgcn_update_dpp(0, __float_as_int(v), 0x140, 0xf, 0xf, true);
    v = fmaxf(v, __int_as_float(t));
    return v;
}
__device__ __forceinline__ float wred32(float v) {
    v = dpp_add16(v);
    v += __shfl_xor(v, 16, 32);   // single cross-half step
    return v;
}
__device__ __forceinline__ float wmax32(float v) {
    v = dpp_max16(v);
    v = fmaxf(v, __shfl_xor(v, 16, 32));
    return v;
}
__device__ __forceinline__ float softplusf(float x) {
    return fmaxf(x, 0.f) + log1pf(expf(-fabsf(x)));
}
// Branchless tanh: sign(x)*(1-e)*rcp(1+e), e = exp(-2|x|).  v_exp_f32 + v_rcp_f32.
__device__ __forceinline__ float fast_tanhf(float x) {
    float ax = fabsf(x);
    float e  = __expf(-2.f * ax);
    float t  = (1.f - e) * __builtin_amdgcn_rcpf(1.f + e);
    return copysignf(t, x);
}

// ---------------- Kernel 0: P = pos_series @ Wp  [S,H] ----------------
__global__ void pos_bias_kernel(const float* __restrict__ pos,
                                const float* __restrict__ Wp,
                                float* __restrict__ P) {
    int idx = blockIdx.x * blockDim.x + threadIdx.x;
    if (idx >= S_LEN * HDIM) return;
    int s = idx / HDIM, h = idx % HDIM;
    const float* pr = pos + s * DDIM;
    float acc = 0.f;
#pragma unroll 4
    for (int d = 0; d < DDIM; ++d) acc += pr[d] * Wp[d * HDIM + h];
    P[idx] = acc;
}

// ---------------- Kernel 1: fused attention + aux terms ----------------
__global__ void __launch_bounds__(256, 2)
fused_attn_kernel(const float* __restrict__ X,       // [B,S,D]
                  const float* __restrict__ Xitem,   // [B,D]
                  const unsigned char* __restrict__ mask, // [B,S] bool
                  const float* __restrict__ origin,  // [B,N,D]
                  const float* __restrict__ We,      // [D,H]
                  const float* __restrict__ z,       // [H]
                  const float* __restrict__ P,       // [S,H]
                  float* __restrict__ out,           // [B,129]
                  float* __restrict__ ws_pos,        // [B]
                  float* __restrict__ ws_neg)        // [B]
{
    extern __shared__ float smem[];
    float* Xs     = smem;                 // S_PAD*DDIM
    float* wsc    = Xs + S_PAD * DDIM;    // 8*S_PAD  (wave-private scores; fully overwritten)
    float* scores = wsc + 8 * S_PAD;      // S_PAD
    float* aovec  = scores + S_PAD;       // DDIM
    float* ovvec  = aovec + DDIM;         // DDIM
    float* lvvec  = ovvec + DDIM;         // DDIM
    float* red    = lvvec + DDIM;         // 16
    int*   ired   = (int*)(red + 16);     // 4 ints

    const int tid  = threadIdx.x;
    const int lane = tid & 31;
    const int wave = tid >> 5;
    const int b    = blockIdx.x;

    // ---- stage X_series[b] into LDS (single global pass), zero pad rows ----
    {
        const float4* src = (const float4*)(X + (size_t)b * S_LEN * DDIM);
        float4* dst = (float4*)Xs;
        for (int i = tid; i < S_LEN * DDIM / 4; i += blockDim.x) dst[i] = src[i];
        for (int i = tid; i < (S_PAD - S_LEN) * DDIM; i += blockDim.x)
            Xs[S_LEN * DDIM + i] = 0.f;
        if (tid == 0) ired[0] = 0;
    }
    __syncthreads();
    if (tid < S_LEN) { if (mask[(size_t)b * S_LEN + tid]) atomicAdd(&ired[0], 1); }

    // ---- WMMA: scores[s] = z . tanh(Xs @ We + P) ----
    // Each wave owns one fixed 16-wide h tile (n0 = wave*16) and register-tiles
    // all 13 s tiles: K is the outer loop, one B fragment per K-step feeds 13
    // independent WMMA accumulation chains.
    const int msel = lane >> 4;   // K-pair selector (ISA A-frag layout)
    const int ml   = lane & 15;
    const int h    = wave * 16 + ml;   // this lane's h column (B-frag N = ml)
    const float zh = z[h];

    v8f acc[MTILES];
#pragma unroll
    for (int mt = 0; mt < MTILES; ++mt)
        acc[mt] = (v8f){0.f, 0.f, 0.f, 0.f, 0.f, 0.f, 0.f, 0.f};

    const float* arow0 = Xs + ml * DDIM;
    for (int kk = 0; kk < 32; ++kk) {
        const int ka = kk * 4 + msel * 2;
        v2f bfr;
        bfr.x = We[ka * HDIM + h];
        bfr.y = We[(ka + 1) * HDIM + h];
#pragma unroll
        for (int mt = 0; mt < MTILES; ++mt) {
            const float* arow = arow0 + mt * 16 * DDIM;
            v2f a; a.x = arow[ka]; a.y = arow[ka + 1];
            acc[mt] = __builtin_amdgcn_wmma_f32_16x16x4_f32(
                false, a, false, bfr, (short)0, acc[mt], false, false);
        }
    }

#pragma unroll
    for (int mt = 0; mt < MTILES; ++mt) {
        const int m0 = mt * 16;
#pragma unroll
        for (int i = 0; i < 8; ++i) {
            const int s = m0 + msel * 8 + i;
            float v = fast_tanhf(acc[mt][i] + P[s * HDIM + h]) * zh;
            v = dpp_add16(v);                       // sum over this wave's 16 h
            if (ml == 0) wsc[wave * S_PAD + s] = v; // each s written exactly once per wave
        }
    }
    __syncthreads();

    // ---- deterministic combine of per-wave partial scores ----
    if (tid < S_PAD) {
        float sc = 0.f;
#pragma unroll
        for (int w = 0; w < 8; ++w) sc += wsc[w * S_PAD + tid];
        scores[tid] = sc;
    }
    __syncthreads();

    const int len  = ired[0];
    const int last = len - 1;

    // ---- masked softmax (wave 0) ----
    if (wave == 0) {
        float m = -INFINITY;
        for (int s = lane; s < len; s += 32) m = fmaxf(m, scores[s]);
        m = wmax32(m);
        float sum = 0.f;
        for (int s = lane; s < len; s += 32) {
            float e = expf(scores[s] - m);
            scores[s] = e; sum += e;
        }
        sum = wred32(sum);
        const float inv = 1.f / sum;
        for (int s = lane; s < S_PAD; s += 32)
            scores[s] = (s < len) ? scores[s] * inv : 0.f;
    }
    __syncthreads();

    // ---- attention_output / out_vec / last_vec per dim (reads LDS only) ----
    if (tid < DDIM) {
        const int d = tid;
        float ao = 0.f, ov = 0.f;
        for (int s = 0; s < len; ++s) {
            float wx = scores[s] * Xs[s * DDIM + d];
            ao += wx;
            if (s < last) ov += wx;
        }
        float lv = scores[last] * Xs[last * DDIM + d];
        aovec[d] = ao; ovvec[d] = ov; lvvec[d] = lv;
        out[(size_t)b * 129 + d] = ao;
    }
    __syncthreads();

    // ---- inner product + positive cosine term (wave 0) ----
    if (wave == 0) {
        float pin = 0.f, plo = 0.f, nl = 0.f, nv = 0.f;
#pragma unroll
        for (int d = lane; d < DDIM; d += 32) {
            float ao = aovec[d], ov = ovvec[d], lv = lvvec[d];
            pin += ao * Xitem[(size_t)b * DDIM + d];
            plo += lv * ov; nl += lv * lv; nv += ov * ov;
        }
        pin = wred32(pin); plo = wred32(plo); nl = wred32(nl); nv = wred32(nv);
        const float ovn = sqrtf(fmaxf(nv, EPSN));
        if (lane == 0) {
            out[(size_t)b * 129 + DDIM] = pin;
            float cosv   = plo / (sqrtf(fmaxf(nl, EPSN)) * ovn);
            float plogit = (1.f - cosv) * 0.5f;
            ws_pos[b] = softplusf(-plogit);
            red[0] = ovn;
        }
    }
    __syncthreads();

    // ---- 64 negatives split across 8 waves ----
    const float ovn = red[0];
    float negacc = 0.f;
    for (int n = wave; n < NNEG; n += 8) {
        const float* org = origin + ((size_t)b * NNEG + n) * DDIM;
        float dp = 0.f, nn = 0.f;
#pragma unroll
        for (int d = lane; d < DDIM; d += 32) {
            float o = org[d];
            dp += o * ovvec[d]; nn += o * o;
        }
        dp = wred32(dp); nn = wred32(nn);
        float cosn = dp / (sqrtf(fmaxf(nn, EPSN)) * ovn);
        negacc += softplusf((1.f - cosn) * 0.5f);
    }
    if (lane == 0) red[2 + wave] = negacc;
    __syncthreads();
    if (tid == 0) {
        float ns = 0.f;
#pragma unroll
        for (int w = 0; w < 8; ++w) ns += red[2 + w];
        ws_neg[b] = ns;
    }
}

// ---------------- Kernel 2: aux[b] = sum_b(pos_loss) + neg_sum[b] ----------------
__global__ void finalize_kernel(const float* __restrict__ ws_pos,
                                const float* __restrict__ ws_neg,
                                float* __restrict__ aux) {
    __shared__ float sred[1024];
    const int t = threadIdx.x;
    sred[t] = ws_pos[t];
    __syncthreads();
    for (int st = 512; st > 0; st >>= 1) {
        if (t < st) sred[t] += sred[t + st];
        __syncthreads();
    }
    aux[t] = sred[0] + ws_neg[t];
}

extern "C" void kernel_launch(void* const* d_in, const int* in_sizes, int n_in,
                              void* d_out, int out_size, void* d_ws, size_t ws_size,
                              hipStream_t stream) {
    const float* X_series   = (const float*)d_in[0];
    const float* pos_series = (const float*)d_in[1];
    const float* X_item     = (const float*)d_in[2];
    const unsigned char* valid_mask = (const unsigned char*)d_in[3];
    const float* origin     = (const float*)d_in[4];
    const float* Wp         = (const float*)d_in[5];
    const float* We         = (const float*)d_in[6];
    const float* z          = (const float*)d_in[7];
    float* out = (float*)d_out;

    float* P      = (float*)d_ws;            // S*H floats
    float* ws_pos = P + S_LEN * HDIM;        // B floats
    float* ws_neg = ws_pos + BATCH;          // B floats

    pos_bias_kernel<<<(S_LEN * HDIM + 255) / 256, 256, 0, stream>>>(pos_series, Wp, P);

    const size_t smem = (size_t)(S_PAD * DDIM + 8 * S_PAD + S_PAD + 3 * DDIM + 16) * sizeof(float)
                      + 4 * sizeof(int);
    fused_attn_kernel<<<BATCH, 256, smem, stream>>>(X_series, X_item, valid_mask, origin,
                                                    We, z, P, out, ws_pos, ws_neg);

    finalize_kernel<<<1, 1024, 0, stream>>>(ws_pos, ws_neg, out + (size_t)BATCH * 129);
}